// CovISqrtHead_7911329759542
// MI455X (gfx1250) — compile-verified
//
#include <hip/hip_runtime.h>
#include <hip/hip_bf16.h>

// ---------------------------------------------------------------------------
// CovISqrtHead on MI455X (gfx1250, wave32, WMMA fp32 path)
//
// Pipeline:
//   k_reduce_gemm : F[b] = w_red(64x512) @ x[b](512x784)        (WMMA f32)
//                   one N-tile per wave, 4 M-accumulators -> each x (B)
//                   fragment is loaded once and reused by 4 WMMAs.
//   k_cov         : cov[b] = Xc Xc^T / (HW-1+1e-5)              (WMMA f32)
//   k_ns          : 5 Newton-Schulz iters, all in 64KB LDS      (WMMA f32)
//   k_proj        : out = cov_sqrt_flat @ w_proj^T + b_proj     (WMMA f32)
//
// Dominant cost: streaming x (103 MB) => ~4.4us at 23.3 TB/s; everything else
// lives in the 192MB L2. fp32 WMMA (16x16x4) keeps full reference precision.
// ---------------------------------------------------------------------------

typedef float v2f __attribute__((ext_vector_type(2)));
typedef float v8f __attribute__((ext_vector_type(8)));

#define WMMA_F32(a, b, c) \
  __builtin_amdgcn_wmma_f32_16x16x4_f32(false, (a), false, (b), (short)0, (c), false, false)

constexpr int BATCH = 64;
constexpr int CIN   = 512;
constexpr int CRED  = 64;
constexpr int HW    = 784;   // 28*28 = 49 tiles of 16
constexpr int PROJ  = 128;
constexpr int CC    = CRED * CRED;  // 4096

// ---------------------------------------------------------------------------
// Kernel 1: F[b][c][n] = sum_k w_red[c][k] * x[b][k][n]
// grid = 64 (batch), block = 256 (8 waves). Each wave owns one of 49 N-tiles
// and accumulates all 4 M-tiles so each x fragment feeds 4 WMMAs.
// ---------------------------------------------------------------------------
__global__ void __launch_bounds__(256)
k_reduce_gemm(const float* __restrict__ x, const float* __restrict__ wred,
              float* __restrict__ F) {
  const int b    = blockIdx.x;
  const int lane = threadIdx.x & 31;
  const int wave = threadIdx.x >> 5;
  const int half = lane >> 4, l16 = lane & 15;

  const float* xb = x + (size_t)b * CIN * HW;
  float*       Fb = F + (size_t)b * CRED * HW;

  for (int nt = wave; nt < HW / 16; nt += 8) {
    v8f acc0 = {}, acc1 = {}, acc2 = {}, acc3 = {};
    const float* xcol = xb + nt * 16 + l16;     // this lane's column of x
    const float* wrow = wred + l16 * CIN;       // this lane's row of w_red

#pragma unroll 4
    for (int kk = 0; kk < CIN; kk += 4) {
      const int k0 = kk + 2 * half;
      v2f bb;
      bb.x = xcol[(size_t)k0 * HW];
      bb.y = xcol[(size_t)(k0 + 1) * HW];
      if ((kk & 15) == 0)  // keep the 23.3 TB/s stream ahead (speculative ok)
        __builtin_prefetch(xcol + (size_t)(k0 + 8) * HW, 0, 1);
      v2f a0, a1, a2, a3;
      a0.x = wrow[ 0 * CIN + k0]; a0.y = wrow[ 0 * CIN + k0 + 1];
      a1.x = wrow[16 * CIN + k0]; a1.y = wrow[16 * CIN + k0 + 1];
      a2.x = wrow[32 * CIN + k0]; a2.y = wrow[32 * CIN + k0 + 1];
      a3.x = wrow[48 * CIN + k0]; a3.y = wrow[48 * CIN + k0 + 1];
      acc0 = WMMA_F32(a0, bb, acc0);
      acc1 = WMMA_F32(a1, bb, acc1);
      acc2 = WMMA_F32(a2, bb, acc2);
      acc3 = WMMA_F32(a3, bb, acc3);
    }

    float* Fo = Fb + nt * 16 + l16;
#pragma unroll
    for (int v = 0; v < 8; ++v) {
      const int r = v + 8 * half;
      Fo[(r +  0) * HW] = acc0[v];
      Fo[(r + 16) * HW] = acc1[v];
      Fo[(r + 32) * HW] = acc2[v];
      Fo[(r + 48) * HW] = acc3[v];
    }
  }
}

// ---------------------------------------------------------------------------
// Kernel 2: per-batch mean-center + covariance. grid = 64, block = 256.
// cov[b] = (F-mu)(F-mu)^T * 1/(HW-1+1e-5).  16 output tiles, 2 per wave.
// ---------------------------------------------------------------------------
__global__ void __launch_bounds__(256)
k_cov(const float* __restrict__ F, float* __restrict__ cov) {
  const int b   = blockIdx.x;
  const int tid = threadIdx.x;
  __shared__ float s_mu[CRED];
  __shared__ float s_part[256];

  const float* Fb = F + (size_t)b * CRED * HW;

  {  // channel means: 4 threads per channel
    const int c = tid >> 2, q = tid & 3;
    float s = 0.f;
    for (int n = q; n < HW; n += 4) s += Fb[c * HW + n];
    s_part[tid] = s;
  }
  __syncthreads();
  if (tid < CRED) {
    s_mu[tid] = (s_part[4 * tid] + s_part[4 * tid + 1] +
                 s_part[4 * tid + 2] + s_part[4 * tid + 3]) / (float)HW;
  }
  __syncthreads();

  const int lane = tid & 31, wave = tid >> 5;
  const int half = lane >> 4, l16 = lane & 15;
  const float inv = 1.0f / ((float)(HW - 1) + 1e-5f);
  float* Cb = cov + (size_t)b * CC;

  for (int t = wave; t < 16; t += 8) {
    const int mt = t >> 2, nt = t & 3;
    const float muA = s_mu[mt * 16 + l16];
    const float muB = s_mu[nt * 16 + l16];
    v8f acc = {};
    for (int kk = 0; kk < HW; kk += 4) {
      const int k0 = kk + 2 * half;
      v2f a, bb;
      a.x  = Fb[(mt * 16 + l16) * HW + k0]     - muA;
      a.y  = Fb[(mt * 16 + l16) * HW + k0 + 1] - muA;
      bb.x = Fb[(nt * 16 + l16) * HW + k0]     - muB;   // Xc^T read
      bb.y = Fb[(nt * 16 + l16) * HW + k0 + 1] - muB;
      acc  = WMMA_F32(a, bb, acc);
    }
#pragma unroll
    for (int v = 0; v < 8; ++v)
      Cb[(mt * 16 + v + 8 * half) * CRED + nt * 16 + l16] = acc[v] * inv;
  }
}

// ---------------------------------------------------------------------------
// 64x64x64 matmul on LDS buffers, 8 waves x 2 tiles each, WMMA f32 16x16x4.
// tmode: D = 1.5*I - 0.5*(A*B)   (the Newton-Schulz T step), else D = A*B.
// ---------------------------------------------------------------------------
__device__ inline void mat64(float* __restrict__ D, const float* __restrict__ A,
                             const float* __restrict__ B, int tid, bool tmode) {
  const int lane = tid & 31, wave = tid >> 5;
  const int half = lane >> 4, l16 = lane & 15;
  for (int t = wave; t < 16; t += 8) {
    const int mt = t >> 2, nt = t & 3;
    v8f acc = {};
#pragma unroll 4
    for (int kk = 0; kk < 64; kk += 4) {
      const int k0 = kk + 2 * half;
      v2f a, bb;
      a.x  = A[(mt * 16 + l16) * 64 + k0];
      a.y  = A[(mt * 16 + l16) * 64 + k0 + 1];
      bb.x = B[k0 * 64 + nt * 16 + l16];
      bb.y = B[(k0 + 1) * 64 + nt * 16 + l16];
      acc  = WMMA_F32(a, bb, acc);
    }
#pragma unroll
    for (int v = 0; v < 8; ++v) {
      const int row = mt * 16 + v + 8 * half, col = nt * 16 + l16;
      float d = acc[v];
      if (tmode) d = (row == col ? 1.5f : 0.0f) - 0.5f * d;
      D[row * 64 + col] = d;
    }
  }
}

// ---------------------------------------------------------------------------
// Kernel 3: Newton-Schulz sqrt. grid = 64 (batch), block = 256.
// 4 x 16KB LDS buffers (Y, Z, T, scratch) selected by integer index so no
// LDS-pointer arrays are ever materialized (avoids addrspacecast initializer).
// ---------------------------------------------------------------------------
__global__ void __launch_bounds__(256)
k_ns(const float* __restrict__ cov, float* __restrict__ out) {
  __shared__ float lds[4 * CC];  // 64 KB of the WGP's 320 KB
  const int b = blockIdx.x, tid = threadIdx.x;
  const float* Ab = cov + (size_t)b * CC;

  // Load A into buf0, accumulate sum of squares into buf2 (free until NS loop)
  float ss = 0.f;
  for (int e = tid; e < CC; e += 256) {
    float v = Ab[e];
    lds[e] = v;
    ss += v * v;
  }
  lds[2 * CC + tid] = ss;
  __syncthreads();
  if (tid == 0) {
    float tot = 0.f;
    for (int i = 0; i < 256; ++i) tot += lds[2 * CC + i];
    lds[2 * CC] = fmaxf(sqrtf(tot), 1e-8f);
  }
  __syncthreads();
  const float nrm  = lds[2 * CC];
  const float rinv = 1.0f / nrm;

  // Y = A/||A||_F in buf0, Z = I in buf1
  for (int e = tid; e < CC; e += 256) {
    lds[e] *= rinv;
    lds[CC + e] = ((e >> 6) == (e & 63)) ? 1.0f : 0.0f;
  }
  __syncthreads();

  int iY = 0, iZ = 1, iT = 2, iS = 3;
  for (int it = 0; it < 5; ++it) {
    mat64(lds + iT * CC, lds + iZ * CC, lds + iY * CC, tid, true);   // T = 1.5I - 0.5*Z*Y
    __syncthreads();
    mat64(lds + iS * CC, lds + iY * CC, lds + iT * CC, tid, false);  // S = Y*T     (new Y)
    __syncthreads();
    mat64(lds + iY * CC, lds + iT * CC, lds + iZ * CC, tid, false);  // oldY <- T*Z (new Z)
    __syncthreads();
    const int nY = iS, nZ = iY, nT = iZ, nS = iT;
    iY = nY; iZ = nZ; iT = nT; iS = nS;
  }

  const float sc = sqrtf(nrm);
  float* Ob = out + (size_t)b * CC;
  const float* Yb = lds + iY * CC;
  for (int e = tid; e < CC; e += 256) Ob[e] = Yb[e] * sc;
}

// ---------------------------------------------------------------------------
// Kernel 4: out[64,128] = cov_sqrt_flat[64,4096] @ w_proj^T + b_proj.
// grid = 4 (batch tiles), block = 256 (8 waves = 8 proj tiles of 16).
// ---------------------------------------------------------------------------
__global__ void __launch_bounds__(256)
k_proj(const float* __restrict__ Ys, const float* __restrict__ wproj,
       const float* __restrict__ bproj, float* __restrict__ out) {
  const int tid  = threadIdx.x;
  const int lane = tid & 31, wave = tid >> 5;
  const int half = lane >> 4, l16 = lane & 15;
  const int mt = blockIdx.x;  // batch tile 0..3
  const int nt = wave;        // proj tile  0..7

  v8f acc = {};
  for (int kk = 0; kk < CC; kk += 4) {
    const int k0 = kk + 2 * half;
    v2f a, bb;
    a.x  = Ys[(size_t)(mt * 16 + l16) * CC + k0];
    a.y  = Ys[(size_t)(mt * 16 + l16) * CC + k0 + 1];
    bb.x = wproj[(size_t)(nt * 16 + l16) * CC + k0];     // w_proj^T
    bb.y = wproj[(size_t)(nt * 16 + l16) * CC + k0 + 1];
    acc  = WMMA_F32(a, bb, acc);
  }
  const float bias = bproj[nt * 16 + l16];
#pragma unroll
  for (int v = 0; v < 8; ++v)
    out[(size_t)(mt * 16 + v + 8 * half) * PROJ + nt * 16 + l16] = acc[v] + bias;
}

// ---------------------------------------------------------------------------
extern "C" void kernel_launch(void* const* d_in, const int* in_sizes, int n_in,
                              void* d_out, int out_size, void* d_ws, size_t ws_size,
                              hipStream_t stream) {
  const float* x     = (const float*)d_in[0];  // [64,512,28,28]
  const float* wred  = (const float*)d_in[1];  // [64,512]
  const float* wproj = (const float*)d_in[2];  // [128,4096]
  const float* bproj = (const float*)d_in[3];  // [128]
  float*       outp  = (float*)d_out;          // [64,128]

  // Workspace layout (floats): F | cov | cov_sqrt  (~15 MB total)
  float* F   = (float*)d_ws;                               // 64*64*784
  float* cov = F   + (size_t)BATCH * CRED * HW;            // 64*4096
  float* cs  = cov + (size_t)BATCH * CC;                   // 64*4096

  k_reduce_gemm<<<dim3(BATCH), dim3(256), 0, stream>>>(x, wred, F);
  k_cov        <<<dim3(BATCH), dim3(256), 0, stream>>>(F, cov);
  k_ns         <<<dim3(BATCH), dim3(256), 0, stream>>>(cov, cs);
  k_proj       <<<dim3(4),     dim3(256), 0, stream>>>(cs, wproj, bproj, outp);
}